// QP_70824010711253
// MI455X (gfx1250) — compile-verified
//
#include <hip/hip_runtime.h>
#include <stdint.h>

namespace {
constexpr int   kS    = 17;     // state dim
constexpr int   kD    = 4;      // action dim
constexpr int   kTPB  = 128;    // envs per block (4 waves, 52KB LDS staging)
constexpr float kEps  = 1e-4f;
constexpr float kSigma = 0.1f;
constexpr int   kIter = 10;
}

// Low 32 bits of a generic pointer into LDS == workgroup-relative LDS byte address
// (shared aperture occupies addr[63:32]; addr[31:0] is the LDS offset).
__device__ __forceinline__ uint32_t lds_lo32(const void* p) {
  return (uint32_t)(uintptr_t)p;
}

__global__ __launch_bounds__(kTPB) void cbf_qp_ipm_kernel(
    const float* __restrict__ u_rl,    // (N,4)
    const float* __restrict__ sdfs,    // (N,1)
    const float* __restrict__ grads,   // (N,17)
    const float* __restrict__ f_drift, // (N,17)
    const float* __restrict__ g_ctrl,  // (N,17,4)
    const float* __restrict__ Lmat,    // (4,4)
    const float* __restrict__ s0,      // (1,)
    float* __restrict__ out)           // (N,4)
{
  __shared__ __align__(16) float sGr[kTPB * kS];          // grads chunk
  __shared__ __align__(16) float sFd[kTPB * kS];          // f_drift chunk
  __shared__ __align__(16) float sGc[kTPB * kS * kD];     // g_ctrl chunk
  __shared__ float sQ[16];
  __shared__ float sQi[16];

  const int tid = threadIdx.x;
  const int e0  = blockIdx.x * kTPB;

  // ---------------- async global -> LDS staging (ASYNCcnt path) -------------
  {
    const uint32_t gr_l = lds_lo32(sGr);
    const uint32_t fd_l = lds_lo32(sFd);
    const uint32_t gc_l = lds_lo32(sGc);
    const uint64_t gr_g = (uint64_t)(uintptr_t)(grads   + (size_t)e0 * kS);
    const uint64_t fd_g = (uint64_t)(uintptr_t)(f_drift + (size_t)e0 * kS);
    const uint64_t gc_g = (uint64_t)(uintptr_t)(g_ctrl  + (size_t)e0 * kS * kD);
    constexpr int n16s = kTPB * kS / 4;       // 544 b128 beats (chunks are 16B aligned: 8704B/blk)
    constexpr int n16g = kTPB * kS * kD / 4;  // 2176 b128 beats (34816B/blk)
    for (int i = tid; i < n16s; i += kTPB) {
      uint32_t la = gr_l + (uint32_t)i * 16u;
      uint64_t ga = gr_g + (uint64_t)i * 16u;
      asm volatile("global_load_async_to_lds_b128 %0, %1, off" :: "v"(la), "v"(ga) : "memory");
      la = fd_l + (uint32_t)i * 16u;
      ga = fd_g + (uint64_t)i * 16u;
      asm volatile("global_load_async_to_lds_b128 %0, %1, off" :: "v"(la), "v"(ga) : "memory");
    }
    for (int i = tid; i < n16g; i += kTPB) {
      const uint32_t la = gc_l + (uint32_t)i * 16u;
      const uint64_t ga = gc_g + (uint64_t)i * 16u;
      asm volatile("global_load_async_to_lds_b128 %0, %1, off" :: "v"(la), "v"(ga) : "memory");
    }
  }

  // Overlap with the async copies: thread 0 builds Q = tril(L) tril(L)^T + eps*I
  // and inverts it (4x4 Gauss-Jordan; Q is SPD, no pivoting needed).
  if (tid == 0) {
    float Lm[4][4];
    #pragma unroll
    for (int r = 0; r < 4; ++r)
      #pragma unroll
      for (int c = 0; c < 4; ++c)
        Lm[r][c] = (c <= r) ? Lmat[r * 4 + c] : 0.0f;
    float Qm[4][4], A[4][4], Inv[4][4];
    #pragma unroll
    for (int r = 0; r < 4; ++r)
      #pragma unroll
      for (int c = 0; c < 4; ++c) {
        float acc = (r == c) ? kEps : 0.0f;
        #pragma unroll
        for (int k = 0; k < 4; ++k) acc += Lm[r][k] * Lm[c][k];
        Qm[r][c] = acc;
        A[r][c] = acc;
        Inv[r][c] = (r == c) ? 1.0f : 0.0f;
      }
    #pragma unroll
    for (int k = 0; k < 4; ++k) {
      const float p = 1.0f / A[k][k];
      #pragma unroll
      for (int c = 0; c < 4; ++c) { A[k][c] *= p; Inv[k][c] *= p; }
      #pragma unroll
      for (int r = 0; r < 4; ++r) {
        if (r == k) continue;
        const float f = A[r][k];
        #pragma unroll
        for (int c = 0; c < 4; ++c) { A[r][c] -= f * A[k][c]; Inv[r][c] -= f * Inv[k][c]; }
      }
    }
    #pragma unroll
    for (int i = 0; i < 16; ++i) { sQ[i] = Qm[i >> 2][i & 3]; sQi[i] = Inv[i >> 2][i & 3]; }
  }

  asm volatile("s_wait_asynccnt 0" ::: "memory");
  __syncthreads();

  // ---------------- per-env setup from LDS (bank-conflict-free, stride 17) --
  const int e = e0 + tid;
  float G[4] = {0.f, 0.f, 0.f, 0.f};
  float gf = 0.f;
  #pragma unroll
  for (int s = 0; s < kS; ++s) {
    const float gs = sGr[tid * kS + s];
    const float fs = sFd[tid * kS + s];
    const float4 gc = *reinterpret_cast<const float4*>(&sGc[(tid * kS + s) * 4]);
    G[0] -= gs * gc.x; G[1] -= gs * gc.y; G[2] -= gs * gc.z; G[3] -= gs * gc.w;
    gf += gs * fs;
  }
  const float4 u = reinterpret_cast<const float4*>(u_rl)[e];
  const float h = s0[0] + sdfs[e] + gf
                - (G[0] * u.x + G[1] * u.y + G[2] * u.z + G[3] * u.w);

  float Q[16], Qi[16];
  #pragma unroll
  for (int i = 0; i < 16; ++i) { Q[i] = sQ[i]; Qi[i] = sQi[i]; }

  // Sherman-Morrison constants: w = Q^-1 G, gQg = G . w
  float w[4];
  #pragma unroll
  for (int a = 0; a < 4; ++a)
    w[a] = Qi[a*4+0]*G[0] + Qi[a*4+1]*G[1] + Qi[a*4+2]*G[2] + Qi[a*4+3]*G[3];
  const float gQg = G[0]*w[0] + G[1]*w[1] + G[2]*w[2] + G[3]*w[3];

  // ---------------- primal-dual IPM, 10 fixed iterations ---------------------
  float z[4] = {0.f, 0.f, 0.f, 0.f};
  float sv = 1.0f, lam = 1.0f;
  #pragma unroll
  for (int it = 0; it < kIter; ++it) {
    const float inv_s = 1.0f / sv;
    const float ratio = lam * inv_s;
    float rx[4];
    #pragma unroll
    for (int a = 0; a < 4; ++a)
      rx[a] = Q[a*4+0]*z[0] + Q[a*4+1]*z[1] + Q[a*4+2]*z[2] + Q[a*4+3]*z[3] + lam * G[a];
    const float gz = G[0]*z[0] + G[1]*z[1] + G[2]*z[2] + G[3]*z[3];
    const float rp = gz + sv - h;
    const float mu = sv * lam;
    const float corr = (kSigma * mu - sv * lam) * inv_s + ratio * rp;
    float rhs[4];
    #pragma unroll
    for (int a = 0; a < 4; ++a) rhs[a] = -(rx[a] + corr * G[a]);
    // dz = H^-1 rhs with H = Q + ratio*G G^T  (Sherman-Morrison on constant Q^-1)
    float t[4];
    #pragma unroll
    for (int a = 0; a < 4; ++a)
      t[a] = Qi[a*4+0]*rhs[0] + Qi[a*4+1]*rhs[1] + Qi[a*4+2]*rhs[2] + Qi[a*4+3]*rhs[3];
    const float gt = G[0]*t[0] + G[1]*t[1] + G[2]*t[2] + G[3]*t[3];
    const float beta = ratio * gt / (1.0f + ratio * gQg);
    float dz[4];
    #pragma unroll
    for (int a = 0; a < 4; ++a) dz[a] = t[a] - beta * w[a];
    const float gdz = G[0]*dz[0] + G[1]*dz[1] + G[2]*dz[2] + G[3]*dz[3];
    const float ds = -rp - gdz;
    const float dlam = (kSigma * mu - sv * lam - lam * ds) * inv_s;
    const float a_s = (ds   < 0.0f) ? (-sv  / ds)   : 1e8f;
    const float a_l = (dlam < 0.0f) ? (-lam / dlam) : 1e8f;
    const float a = 0.99f * fminf(1.0f, fminf(a_s, a_l));
    #pragma unroll
    for (int k = 0; k < 4; ++k) z[k] += a * dz[k];
    sv  += a * ds;
    lam += a * dlam;
  }

  reinterpret_cast<float4*>(out)[e] = make_float4(z[0], z[1], z[2], z[3]);
}

extern "C" void kernel_launch(void* const* d_in, const int* in_sizes, int n_in,
                              void* d_out, int out_size, void* d_ws, size_t ws_size,
                              hipStream_t stream) {
  const float* u_rl    = (const float*)d_in[0];
  const float* sdfs    = (const float*)d_in[1];
  const float* grads   = (const float*)d_in[2];
  const float* f_drift = (const float*)d_in[3];
  const float* g_ctrl  = (const float*)d_in[4];
  const float* Lmat    = (const float*)d_in[5];
  const float* s0      = (const float*)d_in[6];
  float* out = (float*)d_out;

  const int n = in_sizes[1];          // sdfs has N elements
  const int grid = n / kTPB;          // 262144 / 128 = 2048
  cbf_qp_ipm_kernel<<<grid, kTPB, 0, stream>>>(u_rl, sdfs, grads, f_drift,
                                               g_ctrl, Lmat, s0, out);
}